// RBF_PINNs_80711025427082
// MI455X (gfx1250) — compile-verified
//
#include <hip/hip_runtime.h>

// RBF-PINN on gfx1250.
// Transposed WMMA formulation: D = Wpad^T (16x4, rows 0,1 = w0,w1) x phi (4x16, col n = point).
// D rows 0,1 give (u,p) / (lap_u,lap_p) per point-column -> all outputs live in lane registers,
// no LDS staging needed. Accumulation runs on V_WMMA_F32_16X16X4_F32 (fp32-exact), co-executing
// with the VALU/TRANS pipeline that computes r2 and exp.

typedef __attribute__((ext_vector_type(2))) float v2f;
typedef __attribute__((ext_vector_type(8))) float v8f;

#define N_PTS_C 262144
#define N_RBF_C 256
#define WAVES_PER_BLOCK 8
#define PTS_PER_WAVE 16
#define PTS_PER_BLOCK (WAVES_PER_BLOCK * PTS_PER_WAVE) // 128

__constant__ constexpr float kC2 = -0.72134752044448169f; // -0.5*log2(e): exp(-0.5*r2)==exp2(kC2*r2)

// One K=4 chunk: centers [4*i4/... ] -- lane handles centers (2*i4, 2*i4+1) where
// i4 = (k0>>1) + half. Emits two chained WMMAs (phi-matmul and psi-matmul).
__device__ __forceinline__ void rbf_step(const float4* __restrict__ sC4,
                                         const float4* __restrict__ sW4,
                                         int i4, float px, float py,
                                         float s0, float s1,
                                         v8f& accO, v8f& accL)
{
    const float4 cc = sC4[i4];  // cx0, cy0, cx1, cy1
    const float4 wv = sW4[i4];  // w0[c0], w1[c0], w0[c1], w1[c1]

    const float dx0 = px - cc.x, dy0 = py - cc.y;
    const float r20 = fmaf(dx0, dx0, dy0 * dy0);
    const float dx1 = px - cc.z, dy1 = py - cc.w;
    const float r21 = fmaf(dx1, dx1, dy1 * dy1);

    const float phi0 = __builtin_amdgcn_exp2f(kC2 * r20);
    const float phi1 = __builtin_amdgcn_exp2f(kC2 * r21);
    const float psi0 = phi0 * (r20 - 2.0f);   // lap_phi = phi*(4B^2 r2 - 4B) = phi*(r2-2), B=0.5
    const float psi1 = phi1 * (r21 - 2.0f);

    // A (16x4): row m==0 -> w0, row m==1 -> w1, rows >=2 -> 0. Branch-free via float masks.
    v2f Aw; Aw.x = fmaf(wv.y, s1, wv.x * s0);
            Aw.y = fmaf(wv.w, s1, wv.z * s0);
    // B (4x16): row K = 2*half + v, column n = m -> this lane's phi/psi pair.
    v2f Bp; Bp.x = phi0; Bp.y = phi1;
    v2f Bq; Bq.x = psi0; Bq.y = psi1;

    accO = __builtin_amdgcn_wmma_f32_16x16x4_f32(
        false, Aw, false, Bp, (short)0, accO, false, false);
    accL = __builtin_amdgcn_wmma_f32_16x16x4_f32(
        false, Aw, false, Bq, (short)0, accL, false, false);
}

__global__ __launch_bounds__(256) void rbf_pinn_wmma_kernel(
    const float* __restrict__ x, const float* __restrict__ y,
    const float* __restrict__ centers, const float* __restrict__ W,
    const float* __restrict__ eps_p, const float* __restrict__ delta_p,
    const float* __restrict__ gamma_p, float* __restrict__ out)
{
    __shared__ __align__(16) float sC[2 * N_RBF_C]; // (cx, cy) pairs, 2 KB
    __shared__ __align__(16) float sW[2 * N_RBF_C]; // (w0, w1) pairs, 2 KB

    const int tid = threadIdx.x;
    // Preload centers and W into LDS: 128 float4 each, 256 threads -> one b128 apiece.
    if (tid < 128) ((float4*)sC)[tid]       = ((const float4*)centers)[tid];
    else           ((float4*)sW)[tid - 128] = ((const float4*)W)[tid - 128];
    __syncthreads();

    const int wave = tid >> 5;
    const int lane = tid & 31;
    const int m    = lane & 15;   // A-row (W channel select) / B-column (point)
    const int half = lane >> 4;   // selects K offset within chunk

    const int tileBase = (blockIdx.x * WAVES_PER_BLOCK + wave) * PTS_PER_WAVE;
    const int pt = tileBase + m;
    const float px = x[pt];
    const float py = y[pt];

    const float s0 = (m == 0) ? 1.0f : 0.0f;
    const float s1 = (m == 1) ? 1.0f : 0.0f;

    const float4* sC4 = (const float4*)sC;
    const float4* sW4 = (const float4*)sW;

    // Four independent accumulation chains (2 per matmul) to hide WMMA latency.
    v8f accO0 = {}, accL0 = {}, accO1 = {}, accL1 = {};

    #pragma unroll 4
    for (int k0 = 0; k0 < N_RBF_C; k0 += 8) {
        rbf_step(sC4, sW4, (k0 >> 1) + half,     px, py, s0, s1, accO0, accL0);
        rbf_step(sC4, sW4, (k0 >> 1) + 2 + half, px, py, s0, s1, accO1, accL1);
    }

    // D layout: lane = column n (point), VGPR r -> row M = r + 8*half.
    // Lanes 0-15 hold rows 0,1 = (u,p)/(lap_u,lap_p) for point tileBase+lane.
    if (half == 0) {
        const float u  = accO0[0] + accO1[0];
        const float p  = accO0[1] + accO1[1];
        const float lu = accL0[0] + accL1[0];
        const float lp = accL0[1] + accL1[1];

        const float eps   = eps_p[0];
        const float delta = delta_p[0];
        const float gamma = gamma_p[0];

        const float res1 = p - lu;                     // p_pred - lap_u
        const float sh   = u + 2.0f * p + lp;          // u + 2p + lap_p
        const float u2   = u * u;
        const float res2 = eps * u - delta * u2 - gamma * u2 * u - sh;

        float4 o; o.x = u; o.y = p; o.z = res1; o.w = res2;
        ((float4*)out)[pt] = o;                        // (N_PTS, 4) row-major
    }
}

extern "C" void kernel_launch(void* const* d_in, const int* in_sizes, int n_in,
                              void* d_out, int out_size, void* d_ws, size_t ws_size,
                              hipStream_t stream) {
    (void)in_sizes; (void)n_in; (void)d_ws; (void)ws_size; (void)out_size;
    const float* x       = (const float*)d_in[0];
    const float* y       = (const float*)d_in[1];
    const float* centers = (const float*)d_in[2];
    const float* W       = (const float*)d_in[3];
    const float* eps_p   = (const float*)d_in[4];
    const float* delta_p = (const float*)d_in[5];
    const float* gamma_p = (const float*)d_in[6];
    float* out = (float*)d_out;

    const int blocks = N_PTS_C / PTS_PER_BLOCK; // 2048
    rbf_pinn_wmma_kernel<<<blocks, 256, 0, stream>>>(x, y, centers, W,
                                                     eps_p, delta_p, gamma_p, out);
}